// XbmTripletLoss_32298154066255
// MI455X (gfx1250) — compile-verified
//
#include <hip/hip_runtime.h>
#include <hip/hip_bf16.h>

typedef __attribute__((ext_vector_type(2))) float v2f;
typedef __attribute__((ext_vector_type(8))) float v8f;

#define QMAX 32
#define DK 128
#define TILE_N 16
#define TILES_PER_BLOCK 16   // 8 waves x 2 tiles
#define THREADS 256
#define NEG_INF_F (-1e30f)
#define MARGIN_F 0.1f
#define P_MAX 16
#define NN_MAX 32
#define TILE_FLOATS (TILE_N * DK)      // 2048 floats = 8 KB per B tile

// ---- order-preserving float <-> uint map (for atomic max on possibly-negative floats)
__device__ __forceinline__ unsigned f32_ord(float f) {
    unsigned u = __float_as_uint(f);
    return (u & 0x80000000u) ? ~u : (u | 0x80000000u);
}
__device__ __forceinline__ float ord_f32(unsigned u) {
    unsigned v = (u & 0x80000000u) ? (u & 0x7FFFFFFFu) : ~u;
    return __uint_as_float(v);
}

// ---------------------------------------------------------------------------
// Async stage: copy one contiguous 8 KB B tile (16 rows x 128 f32, row-major)
// from global memory into this wave's LDS buffer. 32 lanes x 16 B consecutive
// per instruction -> fully coalesced; tracked on ASYNCcnt (no VGPR data path).
// ---------------------------------------------------------------------------
__device__ __forceinline__ void async_stage_tile(const float* __restrict__ src,
                                                 float* Bbuf, int lane)
{
    unsigned lds_base = (unsigned)(size_t)Bbuf;       // low 32 bits = LDS offset
    unsigned long long gbase = (unsigned long long)(size_t)src;
#pragma unroll
    for (int j = 0; j < 16; ++j) {
        unsigned off = (unsigned)((lane + j * 32) * 16);
        unsigned lds_off = lds_base + off;
        unsigned long long ga = gbase + (unsigned long long)off;
        asm volatile("global_load_async_to_lds_b128 %0, %1, off"
                     :: "v"(lds_off), "v"(ga) : "memory");
    }
}

// ---------------------------------------------------------------------------
// WMMA over one staged tile: C[0:32, n_base:n_base+16] with
// V_WMMA_F32_16X16X4_F32 over K = 128 (two 16x16 accumulators).
// A layout (32-bit A 16x4): lanes 0-15 -> {K0,K1}, lanes 16-31 -> {K2,K3}.
// B layout (32-bit B 4x16): mirrored; b = {Brow[kk], Brow[kk+1]}.
// ---------------------------------------------------------------------------
__device__ __forceinline__ void wmma_from_lds(const float (*A_lds)[DK],
                                              const float* Bbuf, int lane,
                                              v8f& acc0, v8f& acc1)
{
    const int kk_off = (lane >= 16) ? 2 : 0;
    const int m0 = lane & 15;
    const float* brow = Bbuf + (size_t)(lane & 15) * DK;   // row n = lane&15
#pragma unroll 4
    for (int k0 = 0; k0 < DK; k0 += 4) {
        const int kk = k0 + kk_off;
        v2f b  = *(const v2f*)(brow + kk);
        v2f a0 = *(const v2f*)(&A_lds[m0][kk]);
        v2f a1 = *(const v2f*)(&A_lds[m0 + 16][kk]);
        acc0 = __builtin_amdgcn_wmma_f32_16x16x4_f32(false, a0, false, b,
                                                     (short)0, acc0, false, false);
        acc1 = __builtin_amdgcn_wmma_f32_16x16x4_f32(false, a1, false, b,
                                                     (short)0, acc1, false, false);
    }
}

// C/D layout: lane L, vgpr r -> q = r + (L<16 ? 0 : 8), n = n_base + (L&15).
// second accumulator adds +16 to q.

// ---------------------------------------------------------------------------
// Kernel 0: per-column membership bitmasks. posBits[n] bit q = targets_row[n]
// in pidxs[qlocal(q)]; negBits[n] bit q = NOT(targets_row[n] in nnegs[...]).
// ---------------------------------------------------------------------------
__global__ void mask_kernel(const int* __restrict__ targets_row,
                            const int* __restrict__ targets_col,
                            const int* __restrict__ qidxs,
                            const int* __restrict__ pidxs,
                            const int* __restrict__ nnegs,
                            const int* __restrict__ nNegPtr,
                            unsigned* __restrict__ posBits,
                            unsigned* __restrict__ negBits,
                            int Q, int P, int Nn, int m)
{
    __shared__ int pidx_l[QMAX * P_MAX];
    __shared__ int nneg_l[QMAX * NN_MAX];
    const int tid = threadIdx.x;
    const int triplet_len = nNegPtr[0] + 2;
    if (tid < Q) {
        int qid = targets_col[tid * triplet_len];
        int ql = 0;                       // argmax(equality) -> first match, else 0
        for (int j = 0; j < Q; ++j) { if (qidxs[j] == qid) { ql = j; break; } }
        for (int j = 0; j < P;  ++j) pidx_l[tid * P_MAX + j]  = pidxs[ql * P  + j];
        for (int j = 0; j < Nn; ++j) nneg_l[tid * NN_MAX + j] = nnegs[ql * Nn + j];
    }
    __syncthreads();
    const int n = blockIdx.x * THREADS + tid;
    if (n >= m) return;
    const int t = targets_row[n];
    unsigned pb = 0u, nb = 0u;
    for (int q = 0; q < Q; ++q) {
        bool p = false;
        for (int j = 0; j < P; ++j)  p |= (pidx_l[q * P_MAX + j] == t);
        bool nm = false;
        for (int j = 0; j < Nn; ++j) nm |= (nneg_l[q * NN_MAX + j] == t);
        pb |= (p  ? 1u : 0u) << q;
        nb |= (nm ? 0u : 1u) << q;
    }
    posBits[n] = pb;
    negBits[n] = nb;
}

// ---------------------------------------------------------------------------
// Kernel 1: GEMM + masked per-query max (per-block partials).
// ---------------------------------------------------------------------------
__global__ void gemm_max_kernel(const float* __restrict__ inputs_col,
                                const float* __restrict__ inputs_row,
                                const int* __restrict__ nNegPtr,
                                const unsigned* __restrict__ posBits,
                                const unsigned* __restrict__ negBits,
                                float* __restrict__ posMaxPartial,   // [QMAX][NB]
                                float* __restrict__ negMaxPartial,   // [QMAX][NB]
                                int d, int NB)
{
    __shared__ float A_lds[QMAX][DK];                       // 16 KB
    __shared__ float B_all[8][2 * TILE_FLOATS];             // 128 KB (2 bufs/wave)
    __shared__ unsigned redPos[QMAX], redNeg[QMAX];
    const int tid = threadIdx.x;
    const int triplet_len = nNegPtr[0] + 2;
    for (int i = tid; i < QMAX * DK; i += THREADS) {
        int q = i >> 7, k = i & (DK - 1);
        A_lds[q][k] = inputs_col[(size_t)(q * triplet_len) * d + k];
    }
    if (tid < QMAX) { redPos[tid] = 0u; redNeg[tid] = 0u; }   // < ord(NEG_INF)
    __syncthreads();

    const int wave = tid >> 5, lane = tid & 31;
    const int tile0  = blockIdx.x * TILES_PER_BLOCK + wave * 2;
    const int nbase0 = tile0 * TILE_N;
    const int nbase1 = nbase0 + TILE_N;
    float* B0 = &B_all[wave][0];
    float* B1 = &B_all[wave][TILE_FLOATS];

    // double-buffered async DMA: both tiles in flight before first compute
    async_stage_tile(inputs_row + (size_t)nbase0 * d, B0, lane);
    async_stage_tile(inputs_row + (size_t)nbase1 * d, B1, lane);

    v8f accA0 = {}, accA1 = {}, accB0 = {}, accB1 = {};
    asm volatile("s_wait_asynccnt 0x10" ::: "memory");   // tile0 landed (16 still out)
    wmma_from_lds(A_lds, B0, lane, accA0, accA1);
    asm volatile("s_wait_asynccnt 0x0" ::: "memory");    // tile1 landed
    wmma_from_lds(A_lds, B1, lane, accB0, accB1);

    const int qbase = (lane < 16) ? 0 : 8;
#pragma unroll
    for (int t = 0; t < 2; ++t) {
        const int ncol = (t ? nbase1 : nbase0) + (lane & 15);
        const unsigned pb = posBits[ncol], nb = negBits[ncol];
        const v8f& lo = t ? accB0 : accA0;
        const v8f& hi = t ? accB1 : accA1;
#pragma unroll
        for (int r = 0; r < 8; ++r) {
            const float s0 = lo[r], s1 = hi[r];
            const int q0 = qbase + r, q1 = q0 + 16;
            atomicMax(&redPos[q0], f32_ord(((pb >> q0) & 1u) ? s0 : NEG_INF_F));
            atomicMax(&redNeg[q0], f32_ord(((nb >> q0) & 1u) ? s0 : NEG_INF_F));
            atomicMax(&redPos[q1], f32_ord(((pb >> q1) & 1u) ? s1 : NEG_INF_F));
            atomicMax(&redNeg[q1], f32_ord(((nb >> q1) & 1u) ? s1 : NEG_INF_F));
        }
    }
    __syncthreads();
    if (tid < QMAX) {
        posMaxPartial[tid * NB + blockIdx.x] = ord_f32(redPos[tid]);
        negMaxPartial[tid * NB + blockIdx.x] = ord_f32(redNeg[tid]);
    }
}

// ---------------------------------------------------------------------------
// Kernel 2: reduce per-block maxima.
// ---------------------------------------------------------------------------
__global__ void max_reduce_kernel(const float* __restrict__ posMaxPartial,
                                  const float* __restrict__ negMaxPartial,
                                  float* __restrict__ pos_max,
                                  float* __restrict__ neg_max, int NB)
{
    const int q = threadIdx.x;
    if (q >= QMAX) return;
    float pm = NEG_INF_F, nm = NEG_INF_F;
    for (int b = 0; b < NB; ++b) {
        pm = fmaxf(pm, posMaxPartial[q * NB + b]);
        nm = fmaxf(nm, negMaxPartial[q * NB + b]);
    }
    pos_max[q] = pm;
    neg_max[q] = nm;
}

// ---------------------------------------------------------------------------
// Kernel 3: GEMM again + selection sums/counts (per-block partials).
// ---------------------------------------------------------------------------
__global__ void gemm_sel_kernel(const float* __restrict__ inputs_col,
                                const float* __restrict__ inputs_row,
                                const int* __restrict__ nNegPtr,
                                const unsigned* __restrict__ posBits,
                                const unsigned* __restrict__ negBits,
                                const float* __restrict__ pos_max,
                                const float* __restrict__ neg_max,
                                float* __restrict__ posSumPartial,   // [QMAX][NB]
                                float* __restrict__ negSumPartial,
                                int*   __restrict__ posCntPartial,
                                int*   __restrict__ negCntPartial,
                                int d, int NB)
{
    __shared__ float A_lds[QMAX][DK];
    __shared__ float B_all[8][2 * TILE_FLOATS];
    __shared__ float sSumP[QMAX], sSumN[QMAX];
    __shared__ int   sCntP[QMAX], sCntN[QMAX];
    __shared__ float thrPos[QMAX], thrNeg[QMAX];
    const int tid = threadIdx.x;
    const int triplet_len = nNegPtr[0] + 2;
    for (int i = tid; i < QMAX * DK; i += THREADS) {
        int q = i >> 7, k = i & (DK - 1);
        A_lds[q][k] = inputs_col[(size_t)(q * triplet_len) * d + k];
    }
    if (tid < QMAX) {
        sSumP[tid] = 0.0f; sSumN[tid] = 0.0f; sCntP[tid] = 0; sCntN[tid] = 0;
        thrNeg[tid] = neg_max[tid] + MARGIN_F;                 // pos_sel: s < this
        thrPos[tid] = fmaxf(0.4f, pos_max[tid]) - MARGIN_F;    // neg_sel: s > this
    }
    __syncthreads();

    const int wave = tid >> 5, lane = tid & 31;
    const int tile0  = blockIdx.x * TILES_PER_BLOCK + wave * 2;
    const int nbase0 = tile0 * TILE_N;
    const int nbase1 = nbase0 + TILE_N;
    float* B0 = &B_all[wave][0];
    float* B1 = &B_all[wave][TILE_FLOATS];

    async_stage_tile(inputs_row + (size_t)nbase0 * d, B0, lane);
    async_stage_tile(inputs_row + (size_t)nbase1 * d, B1, lane);

    v8f accA0 = {}, accA1 = {}, accB0 = {}, accB1 = {};
    asm volatile("s_wait_asynccnt 0x10" ::: "memory");
    wmma_from_lds(A_lds, B0, lane, accA0, accA1);
    asm volatile("s_wait_asynccnt 0x0" ::: "memory");
    wmma_from_lds(A_lds, B1, lane, accB0, accB1);

    const int qbase = (lane < 16) ? 0 : 8;
#pragma unroll
    for (int t = 0; t < 2; ++t) {
        const int ncol = (t ? nbase1 : nbase0) + (lane & 15);
        const unsigned pb = posBits[ncol], nb = negBits[ncol];
        const v8f& lo = t ? accB0 : accA0;
        const v8f& hi = t ? accB1 : accA1;
#pragma unroll
        for (int r = 0; r < 8; ++r) {
#pragma unroll
            for (int half = 0; half < 2; ++half) {
                const float s = half ? hi[r] : lo[r];
                const int q = qbase + r + (half ? 16 : 0);
                if (((pb >> q) & 1u) && (s < thrNeg[q])) {
                    atomicAdd(&sSumP[q], 1.0f - s);
                    atomicAdd(&sCntP[q], 1);
                }
                if (((nb >> q) & 1u) && (s > thrPos[q])) {
                    atomicAdd(&sSumN[q], s);
                    atomicAdd(&sCntN[q], 1);
                }
            }
        }
    }
    __syncthreads();
    if (tid < QMAX) {
        posSumPartial[tid * NB + blockIdx.x] = sSumP[tid];
        negSumPartial[tid * NB + blockIdx.x] = sSumN[tid];
        posCntPartial[tid * NB + blockIdx.x] = sCntP[tid];
        negCntPartial[tid * NB + blockIdx.x] = sCntN[tid];
    }
}

// ---------------------------------------------------------------------------
// Kernel 4: final scalar loss.
// ---------------------------------------------------------------------------
__global__ void final_kernel(const float* __restrict__ posSumPartial,
                             const float* __restrict__ negSumPartial,
                             const int*   __restrict__ posCntPartial,
                             const int*   __restrict__ negCntPartial,
                             const float* __restrict__ pos_max,
                             float* __restrict__ out, int Q, int NB)
{
    __shared__ float contrib[QMAX];
    const int q = threadIdx.x;
    if (q < QMAX) {
        float sp = 0.0f, sn = 0.0f;
        int cp = 0, cn = 0;
        for (int b = 0; b < NB; ++b) {
            sp += posSumPartial[q * NB + b];
            sn += negSumPartial[q * NB + b];
            cp += posCntPartial[q * NB + b];
            cn += negCntPartial[q * NB + b];
        }
        const float pl = (cp > 0) ? sp / (float)cp : 0.0f;
        const float nl = (cn > 0) ? sn / (float)cn : 0.0f;
        const bool has_pos = pos_max[q] > -1e29f;   // any positive exists
        contrib[q] = has_pos ? (pl + nl) : 0.0f;
    }
    __syncthreads();
    if (q == 0) {
        float tot = 0.0f;
        for (int i = 0; i < Q; ++i) tot += contrib[i];
        out[0] = tot / (float)Q;
    }
}

// ---------------------------------------------------------------------------
extern "C" void kernel_launch(void* const* d_in, const int* in_sizes, int n_in,
                              void* d_out, int out_size, void* d_ws, size_t ws_size,
                              hipStream_t stream)
{
    const float* inputs_col  = (const float*)d_in[0];
    const float* inputs_row  = (const float*)d_in[1];
    const int*   targets_col = (const int*)d_in[2];
    const int*   targets_row = (const int*)d_in[3];
    const int*   qidxs       = (const int*)d_in[4];
    const int*   pidxs       = (const int*)d_in[5];
    const int*   nnegs       = (const int*)d_in[6];
    const int*   nNegPtr     = (const int*)d_in[7];

    const int n  = in_sizes[2];
    const int m  = in_sizes[3];
    const int Q  = in_sizes[4];
    const int d  = in_sizes[0] / n;          // 128
    const int P  = in_sizes[5] / Q;          // 10
    const int Nn = in_sizes[6] / Q;          // 25

    const int COLS_PER_BLOCK = TILES_PER_BLOCK * TILE_N;   // 256
    const int NB = m / COLS_PER_BLOCK;                     // 256

    // workspace layout
    char* ws = (char*)d_ws;
    unsigned* posBits       = (unsigned*)ws; ws += (size_t)m * 4;
    unsigned* negBits       = (unsigned*)ws; ws += (size_t)m * 4;
    float* posMaxPartial    = (float*)ws;    ws += (size_t)QMAX * NB * 4;
    float* negMaxPartial    = (float*)ws;    ws += (size_t)QMAX * NB * 4;
    float* posSumPartial    = (float*)ws;    ws += (size_t)QMAX * NB * 4;
    float* negSumPartial    = (float*)ws;    ws += (size_t)QMAX * NB * 4;
    int*   posCntPartial    = (int*)ws;      ws += (size_t)QMAX * NB * 4;
    int*   negCntPartial    = (int*)ws;      ws += (size_t)QMAX * NB * 4;
    float* pos_max          = (float*)ws;    ws += (size_t)QMAX * 4;
    float* neg_max          = (float*)ws;    ws += (size_t)QMAX * 4;
    float* out = (float*)d_out;

    mask_kernel<<<(m + THREADS - 1) / THREADS, THREADS, 0, stream>>>(
        targets_row, targets_col, qidxs, pidxs, nnegs, nNegPtr,
        posBits, negBits, Q, P, Nn, m);

    gemm_max_kernel<<<NB, THREADS, 0, stream>>>(
        inputs_col, inputs_row, nNegPtr, posBits, negBits,
        posMaxPartial, negMaxPartial, d, NB);

    max_reduce_kernel<<<1, 32, 0, stream>>>(
        posMaxPartial, negMaxPartial, pos_max, neg_max, NB);

    gemm_sel_kernel<<<NB, THREADS, 0, stream>>>(
        inputs_col, inputs_row, nNegPtr, posBits, negBits,
        pos_max, neg_max,
        posSumPartial, negSumPartial, posCntPartial, negCntPartial, d, NB);

    final_kernel<<<1, 32, 0, stream>>>(
        posSumPartial, negSumPartial, posCntPartial, negCntPartial,
        pos_max, out, Q, NB);
}